// JLFisherRegularizer_39367670235337
// MI455X (gfx1250) — compile-verified
//
#include <hip/hip_runtime.h>

typedef float v2f __attribute__((ext_vector_type(2)));
typedef float v8f __attribute__((ext_vector_type(8)));

#define PDIM   2048
#define BATCH  8192
#define KT     64                 // K-chunk staged in LDS per double-buffer slot
#define NCHUNK (BATCH / KT)       // 128
#define MTI    64                 // macro-tile rows  (i)
#define MTJ    128                // macro-tile cols  (j)

// ---------------------------------------------------------------------------
// CDNA5 async global->LDS copy (ASYNCcnt), GVS form: saddr pair + 32-bit voffset.
// ---------------------------------------------------------------------------
__device__ __forceinline__ void async_ld_b128(unsigned lds_off, unsigned voff,
                                              const float* base) {
    asm volatile("global_load_async_to_lds_b128 %0, %1, %2"
                 :: "v"(lds_off), "v"(voff), "s"(base) : "memory");
}
__device__ __forceinline__ void wait_async_le12() {
    asm volatile("s_wait_asynccnt 0xC" ::: "memory");
}
__device__ __forceinline__ void wait_async_0() {
    asm volatile("s_wait_asynccnt 0x0" ::: "memory");
}

// ---------------------------------------------------------------------------
// Kernel 1: S = (G^T G)/B.  64x128 macro-tile per 256-thread block (8 waves).
// Each wave: 2 A-fragments x 2 B-fragments = four 16x16 WMMA tiles
// (1 ds_load_2addr_stride64 per WMMA).  G panels double-buffered in LDS via
// global_load_async_to_lds_b128; f64 fold every KT=64 K-values.
// ---------------------------------------------------------------------------
__global__ void __launch_bounds__(256)
syrk_tile_wmma(const float* __restrict__ G, float* __restrict__ S) {
    __shared__ float sA[2 * KT * MTI];   // [buf][k][64]  cols Ibase..+63
    __shared__ float sB[2 * KT * MTJ];   // [buf][k][128] cols Jbase..+127

    const int tim = blockIdx.x;          // macro row  0..31  (64-row strip)
    const int tjm = blockIdx.y;          // macro col  0..15  (128-col strip)
    if (2 * tjm > tim) return;           // lower-triangle coverage (whole block)

    const int t    = threadIdx.x;
    const int lane = t & 31;
    const int wave = t >> 5;             // 0..7
    const int si   = (wave >> 2) * 16;   // A sub-col: 0/16  (+32 for 2nd frag)
    const int sjb  = (wave & 3) * 16;    // B sub-col: 0,16,32,48 (+64 for 2nd frag)
    const int n16  = lane & 15;
    const int kb   = (lane >> 4) << 1;   // K base within 4: 0 / 2

    const int Ibase = tim * MTI;
    const int Jbase = tjm * MTJ;

    const unsigned ldsA0 = (unsigned)(size_t)(void*)sA;
    const unsigned ldsB0 = (unsigned)(size_t)(void*)sB;

    // Opaque inter-fragment byte offsets: prevents cross-fragment ds_load
    // fusion so each fragment lowers to one ds_load_2addr_stride64_b32
    // straight into an even-aligned VGPR pair (no re-pairing moves).
    unsigned offA = 32u * 4u;            // A fragment separation (bytes)
    unsigned offB = 64u * 4u;            // B fragment separation (bytes)
    asm("" : "+v"(offA));
    asm("" : "+v"(offB));

    // async staging maps: A: 16 lanes x b128 per 64-float row; B: 32 lanes per 128-float row
    const unsigned rowA = (unsigned)(t >> 4), colA = (unsigned)((t & 15) * 4);
    const unsigned rowB = (unsigned)(t >> 5), colB = (unsigned)((t & 31) * 4);

    // prologue: stage chunk 0 into buffer 0
    {
#pragma unroll
        for (int p = 0; p < 4; ++p) {
            const unsigned r = rowA + (unsigned)(p * 16);
            async_ld_b128(ldsA0 + (r * MTI + colA) * 4u,
                          (r * PDIM + (unsigned)Ibase + colA) * 4u, G);
        }
#pragma unroll
        for (int p = 0; p < 8; ++p) {
            const unsigned r = rowB + (unsigned)(p * 8);
            async_ld_b128(ldsB0 + (r * MTJ + colB) * 4u,
                          (r * PDIM + (unsigned)Jbase + colB) * 4u, G);
        }
    }

    double d00[8] = {0,0,0,0,0,0,0,0};
    double d01[8] = {0,0,0,0,0,0,0,0};
    double d10[8] = {0,0,0,0,0,0,0,0};
    double d11[8] = {0,0,0,0,0,0,0,0};

    for (int c = 0; c < NCHUNK; ++c) {
        const int buf = c & 1;

        // issue next chunk into the other buffer (freed by barrier at end of c-1)
        if (c + 1 < NCHUNK) {
            const unsigned kn  = (unsigned)((c + 1) * KT);
            const unsigned ba  = (unsigned)((c + 1) & 1) * (KT * MTI * 4u);
            const unsigned bb  = (unsigned)((c + 1) & 1) * (KT * MTJ * 4u);
#pragma unroll
            for (int p = 0; p < 4; ++p) {
                const unsigned r = rowA + (unsigned)(p * 16);
                async_ld_b128(ldsA0 + ba + (r * MTI + colA) * 4u,
                              ((kn + r) * PDIM + (unsigned)Ibase + colA) * 4u, G);
            }
#pragma unroll
            for (int p = 0; p < 8; ++p) {
                const unsigned r = rowB + (unsigned)(p * 8);
                async_ld_b128(ldsB0 + bb + (r * MTJ + colB) * 4u,
                              ((kn + r) * PDIM + (unsigned)Jbase + colB) * 4u, G);
            }
            wait_async_le12();   // chunk c's 12 transfers complete (in-order)
        } else {
            wait_async_0();
        }
        __syncthreads();

        // consume buffer `buf`: 16 K-steps x 4 WMMAs, per-fragment base pointers
        v8f a00 = {0,0,0,0,0,0,0,0}, a01 = {0,0,0,0,0,0,0,0};
        v8f a10 = {0,0,0,0,0,0,0,0}, a11 = {0,0,0,0,0,0,0,0};
        const float* a0p = sA + buf * (KT * MTI) + si + n16;                 // A frag 0
        const float* a1p = (const float*)((const char*)a0p + offA);          // A frag 1
        const float* b0p = sB + buf * (KT * MTJ) + sjb + n16;                // B frag 0
        const float* b1p = (const float*)((const char*)b0p + offB);          // B frag 1
#pragma unroll
        for (int ks = 0; ks < KT / 4; ++ks) {
            const int kr = ks * 4 + kb;
            v2f fa0, fa1, fb0, fb1;
            fa0.x = a0p[kr * MTI];  fa0.y = a0p[kr * MTI + MTI];
            fa1.x = a1p[kr * MTI];  fa1.y = a1p[kr * MTI + MTI];
            fb0.x = b0p[kr * MTJ];  fb0.y = b0p[kr * MTJ + MTJ];
            fb1.x = b1p[kr * MTJ];  fb1.y = b1p[kr * MTJ + MTJ];
            a00 = __builtin_amdgcn_wmma_f32_16x16x4_f32(false, fa0, false, fb0, (short)0, a00, false, false);
            a01 = __builtin_amdgcn_wmma_f32_16x16x4_f32(false, fa0, false, fb1, (short)0, a01, false, false);
            a10 = __builtin_amdgcn_wmma_f32_16x16x4_f32(false, fa1, false, fb0, (short)0, a10, false, false);
            a11 = __builtin_amdgcn_wmma_f32_16x16x4_f32(false, fa1, false, fb1, (short)0, a11, false, false);
        }
#pragma unroll
        for (int v = 0; v < 8; ++v) {
            d00[v] += (double)a00[v];
            d01[v] += (double)a01[v];
            d10[v] += (double)a10[v];
            d11[v] += (double)a11[v];
        }
        __syncthreads();   // all waves done with `buf` before it is overwritten
    }

    // write-out: D layout VGPR v -> M = v (+8 for lanes 16-31); dual store => symmetry
    const double invB = 1.0 / (double)BATCH;
    const int mofs = (lane >> 4) << 3;
#pragma unroll
    for (int v = 0; v < 8; ++v) {
        const int m  = v + mofs;
        const int i0 = Ibase + si + m,        i1 = Ibase + si + 32 + m;
        const int j0 = Jbase + sjb + n16,     j1 = Jbase + sjb + 64 + n16;
        float w00 = (float)(d00[v] * invB);
        float w01 = (float)(d01[v] * invB);
        float w10 = (float)(d10[v] * invB);
        float w11 = (float)(d11[v] * invB);
        S[(size_t)i0 * PDIM + j0] = w00;  S[(size_t)j0 * PDIM + i0] = w00;
        S[(size_t)i0 * PDIM + j1] = w01;  S[(size_t)j1 * PDIM + i0] = w01;
        S[(size_t)i1 * PDIM + j0] = w10;  S[(size_t)j0 * PDIM + i1] = w10;
        S[(size_t)i1 * PDIM + j1] = w11;  S[(size_t)j1 * PDIM + i1] = w11;
    }
}

// ---------------------------------------------------------------------------
// Kernel 2: y = alpha*x + beta*(A x), f64 accumulation, one wave per row.
// ---------------------------------------------------------------------------
__global__ void matvec_f64(const float* __restrict__ A, const double* __restrict__ x,
                           double* __restrict__ y, const double* __restrict__ scal,
                           int shift_slot, double beta) {
    const int lane = threadIdx.x & 31;
    const int row  = (int)((blockIdx.x * blockDim.x + threadIdx.x) >> 5);
    if (row >= PDIM) return;

    const float* Ar = A + (size_t)row * PDIM;
    double s = 0.0;
    for (int c = lane; c < PDIM; c += 32)
        s += (double)Ar[c] * x[c];
#pragma unroll
    for (int off = 16; off > 0; off >>= 1)
        s += __shfl_xor(s, off, 32);
    if (lane == 0) {
        const double alpha = (shift_slot >= 0) ? scal[shift_slot] : 0.0;
        y[row] = alpha * x[row] + beta * s;
    }
}

// ---------------------------------------------------------------------------
// Kernel 3: Rayleigh quotient ray = x.y, then x <- y/||y||.  One 256-thr block.
// ---------------------------------------------------------------------------
__global__ void normalize_rayleigh(double* __restrict__ x, const double* __restrict__ y,
                                   double* __restrict__ scal, int ray_slot) {
    __shared__ double sdot[256];
    __shared__ double snrm[256];
    __shared__ double sinv;
    const int t = threadIdx.x;

    double yv[PDIM / 256];
    double pd = 0.0, pn = 0.0;
#pragma unroll
    for (int i = 0; i < PDIM / 256; ++i) {
        const int idx = t + i * 256;
        const double yy = y[idx];
        yv[i] = yy;
        pd += x[idx] * yy;
        pn += yy * yy;
    }
    sdot[t] = pd;
    snrm[t] = pn;
    __syncthreads();
    for (int s = 128; s > 0; s >>= 1) {
        if (t < s) { sdot[t] += sdot[t + s]; snrm[t] += snrm[t + s]; }
        __syncthreads();
    }
    if (t == 0) {
        scal[ray_slot] = sdot[0];
        sinv = rsqrt(fmax(snrm[0], 1e-300));
    }
    __syncthreads();
    const double inv = sinv;
#pragma unroll
    for (int i = 0; i < PDIM / 256; ++i)
        x[t + i * 256] = yv[i] * inv;
}

// ---------------------------------------------------------------------------
// Helpers.
// ---------------------------------------------------------------------------
__global__ void init_x(double* __restrict__ x, unsigned seed) {
    const int i = (int)(blockIdx.x * blockDim.x + threadIdx.x);
    if (i < PDIM) {
        unsigned h = (unsigned)i * 2654435761u + seed;
        h ^= h >> 16; h *= 2246822519u; h ^= h >> 13;
        x[i] = ((double)(h & 0xFFFFFFu) / 16777216.0) - 0.5;
    }
}

__global__ void set_shift(double* __restrict__ scal) {
    if (threadIdx.x == 0 && blockIdx.x == 0)
        scal[1] = scal[0] * 1.01 + 1e-9;     // s slightly above lambda_max
}

__global__ void finalize(const double* __restrict__ scal, float* __restrict__ out) {
    if (threadIdx.x == 0 && blockIdx.x == 0) {
        const double lam_min = scal[1] - scal[2];          // s - mu_max(sI - A)
        const double pen     = fmax(0.0, 1.0 - lam_min);   // relu(DELTA - lambda_1)
        out[0] = (float)(0.1 * pen);                       // SPECTRAL_WEIGHT
    }
}

// ---------------------------------------------------------------------------
// Orchestration (graph-capture safe, all on `stream`).
// ws: [ fisher f32 16MiB | x f64 | y f64 | scal f64 ]
// ---------------------------------------------------------------------------
extern "C" void kernel_launch(void* const* d_in, const int* in_sizes, int n_in,
                              void* d_out, int out_size, void* d_ws, size_t ws_size,
                              hipStream_t stream) {
    (void)in_sizes; (void)n_in; (void)out_size; (void)ws_size;

    const float* G = (const float*)d_in[0];
    char*   w      = (char*)d_ws;
    float*  fisher = (float*)w;
    double* x      = (double*)(w + (size_t)PDIM * PDIM * sizeof(float));
    double* y      = x + PDIM;
    double* scal   = y + PDIM;

    // 1) Fisher = G^T G / B (WMMA f32 + async LDS double buffering)
    dim3 sgrid(PDIM / MTI, PDIM / MTJ);           // 32 x 16
    syrk_tile_wmma<<<sgrid, 256, 0, stream>>>(G, fisher);

    // 2) lambda_max via power iteration (Rayleigh in scal[0])
    init_x<<<(PDIM + 255) / 256, 256, 0, stream>>>(x, 12345u);
    for (int it = 0; it < 32; ++it) {
        matvec_f64<<<PDIM / 8, 256, 0, stream>>>(fisher, x, y, scal, -1, 1.0);
        normalize_rayleigh<<<1, 256, 0, stream>>>(x, y, scal, 0);
    }
    set_shift<<<1, 32, 0, stream>>>(scal);        // scal[1] = s

    // 3) lambda_min via power iteration on (s*I - A)
    init_x<<<(PDIM + 255) / 256, 256, 0, stream>>>(x, 98765u);
    for (int it = 0; it < 96; ++it) {
        matvec_f64<<<PDIM / 8, 256, 0, stream>>>(fisher, x, y, scal, 1, -1.0);
        normalize_rayleigh<<<1, 256, 0, stream>>>(x, y, scal, 2);
    }

    // 4) out = 0.1 * relu(1 - lambda_min)
    finalize<<<1, 32, 0, stream>>>(scal, (float*)d_out);
}